// ssh_deformation_dirichlet_82592221102718
// MI455X (gfx1250) — compile-verified
//
#include <hip/hip_runtime.h>
#include <hip/hip_bf16.h>
#include <cstdint>
#include <math.h>

#define MESH 4194304
#define BLK  256
#define NWAVE (BLK / 32)

// ---------------------------------------------------------------------------
// gfx1250 async global->LDS path (probe via __has_builtin; fallback = plain LDS)
// Builtin signatures (confirmed by hipcc):
//   b32: (int AS1*, int AS3*, imm int, imm int)
//   b64: (v2i AS1*, v2i AS3*, imm int, imm int)
// ---------------------------------------------------------------------------
#if defined(__has_builtin)
#  if __has_builtin(__builtin_amdgcn_global_load_async_to_lds_b64) && \
      __has_builtin(__builtin_amdgcn_global_load_async_to_lds_b32)
#    define ASYNC_LDS 1
#  endif
#endif

#ifdef ASYNC_LDS
typedef int v2i __attribute__((ext_vector_type(2)));
typedef __attribute__((address_space(1))) v2i g_v2i;
typedef __attribute__((address_space(3))) v2i l_v2i;
typedef __attribute__((address_space(1))) int g_i32;
typedef __attribute__((address_space(3))) int l_i32;
#  if __has_builtin(__builtin_amdgcn_s_wait_asynccnt)
#    define WAIT_ASYNC() __builtin_amdgcn_s_wait_asynccnt(0)
#  else
#    define WAIT_ASYNC() asm volatile("s_wait_asynccnt 0" ::: "memory")
#  endif
#endif

// ws layout (doubles): [0]=loss_v  [1]=loss_s  [2]=loss_so2

// Load a tile of state[i0 .. i0+BLK] (halo wraps mod MESH) and
// theta[i0-1 .. i0+BLK-1] (out-of-range slots -> 0) into LDS.
__device__ __forceinline__ void load_tile(const float* __restrict__ theta,
                                          const float2* __restrict__ state,
                                          int i0, float2* s_state, float* s_theta) {
    const int tid = threadIdx.x;
    const int g   = i0 + tid;                 // always < MESH
    const int tg  = i0 - 1 + tid;             // theta index for slot tid
#ifdef ASYNC_LDS
    __builtin_amdgcn_global_load_async_to_lds_b64((g_v2i*)(state + g),
                                                  (l_v2i*)(s_state + tid), 0, 0);
    if (tg >= 0 && tg <= MESH - 2) {
        __builtin_amdgcn_global_load_async_to_lds_b32((g_i32*)(theta + tg),
                                                      (l_i32*)(s_theta + tid), 0, 0);
    } else {
        s_theta[tid] = 0.0f;
    }
    if (tid == 0) {
        int h = i0 + BLK; if (h >= MESH) h -= MESH;   // halo (wraps to 0 on last block)
        __builtin_amdgcn_global_load_async_to_lds_b64((g_v2i*)(state + h),
                                                      (l_v2i*)(s_state + BLK), 0, 0);
        int hg = i0 + BLK - 1;
        if (hg <= MESH - 2) {
            __builtin_amdgcn_global_load_async_to_lds_b32((g_i32*)(theta + hg),
                                                          (l_i32*)(s_theta + BLK), 0, 0);
        } else {
            s_theta[BLK] = 0.0f;
        }
    }
    WAIT_ASYNC();
#else
    s_state[tid] = state[g];
    s_theta[tid] = (tg >= 0 && tg <= MESH - 2) ? theta[tg] : 0.0f;
    if (tid == 0) {
        int h = i0 + BLK; if (h >= MESH) h -= MESH;
        s_state[BLK] = state[h];
        int hg = i0 + BLK - 1;
        s_theta[BLK] = (hg <= MESH - 2) ? theta[hg] : 0.0f;
    }
#endif
    __syncthreads();
}

// deformed(i): state[0] untouched, otherwise rotate by theta[i-1]
__device__ __forceinline__ float2 make_deformed(int i, float2 st, float th) {
    if (i == 0) return st;
    float c = __cosf(th), s = __sinf(th);
    return make_float2(c * st.x - s * st.y, s * st.x + c * st.y);
}

// Shared per-block pipeline: build deformed tile in LDS, return this thread's vol.
__device__ __forceinline__ float tile_vol(int i0, const float2* s_state,
                                          const float* s_theta, float2* s_df) {
    const int tid = threadIdx.x;
    s_df[tid] = make_deformed(i0 + tid, s_state[tid], s_theta[tid]);
    if (tid == 0) {
        int h = i0 + BLK; if (h >= MESH) h -= MESH;
        s_df[BLK] = make_deformed(h, s_state[BLK], s_theta[BLK]);
    }
    __syncthreads();
    float2 a = s_df[tid], b = s_df[tid + 1];
    float dot = a.x * b.x + a.y * b.y;
    return sqrtf(fabsf(1.0f - dot * dot));
}

// wave32 butterfly reduction (5 shuffle steps, no LDS, no barriers)
__device__ __forceinline__ float wave_reduce(float v) {
    #pragma unroll
    for (int m = 16; m > 0; m >>= 1) v += __shfl_xor(v, m, 32);
    return v;
}

// Block reduce of two values: per-wave shuffle, one barrier, wave0 folds 8 partials.
__device__ __forceinline__ void block_reduce2(float v0, float v1,
                                              float* s_w0, float* s_w1,
                                              double* d0, double* d1) {
    const int tid  = threadIdx.x;
    const int wave = tid >> 5;
    const int lane = tid & 31;
    float r0 = wave_reduce(v0);
    float r1 = d1 ? wave_reduce(v1) : 0.0f;
    if (lane == 0) { s_w0[wave] = r0; if (d1) s_w1[wave] = r1; }
    __syncthreads();
    if (wave == 0) {
        float x0 = (lane < NWAVE) ? s_w0[lane] : 0.0f;
        x0 = wave_reduce(x0);
        if (d1) {
            float x1 = (lane < NWAVE) ? s_w1[lane] : 0.0f;
            x1 = wave_reduce(x1);
            if (lane == 0) atomicAdd(d1, (double)x1);
        }
        if (lane == 0) atomicAdd(d0, (double)x0);
    }
}

__global__ void ssh_pass1(const float* __restrict__ theta,
                          const float2* __restrict__ state,
                          double* __restrict__ ws) {
    __shared__ float2 s_state[BLK + 1];
    __shared__ float  s_theta[BLK + 1];
    __shared__ float2 s_df[BLK + 1];
    __shared__ float  s_w0[NWAVE];
    __shared__ float  s_w1[NWAVE];
    const int i0 = blockIdx.x * BLK;
    load_tile(theta, state, i0, s_state, s_theta);
    float vol = tile_vol(i0, s_state, s_theta, s_df);
    block_reduce2(vol, vol * vol, s_w0, s_w1, &ws[0], &ws[1]);
}

__global__ void ssh_pass2(const float* __restrict__ theta,
                          const float2* __restrict__ state,
                          double* __restrict__ ws) {
    __shared__ float2 s_state[BLK + 1];
    __shared__ float  s_theta[BLK + 1];
    __shared__ float2 s_df[BLK + 1];
    __shared__ float  s_w0[NWAVE];
    const int i0  = blockIdx.x * BLK;
    const int tid = threadIdx.x;
    const float avg_len = (float)(fabs(ws[0]) / (double)MESH);
    load_tile(theta, state, i0, s_state, s_theta);
    float vol = tile_vol(i0, s_state, s_theta, s_df);

    // weights: j==0 -> |theta[0]|*exp(-vol/avg)
    //          1<=j<=M-2 -> |theta[j-1]-theta[j]|*exp(-vol/avg)
    //          j==M-1 -> |theta[M-2]|*exp(-vol^2/avg)   (sim_last uses 1-dots^2)
    const int j = i0 + tid;
    float w, arg;
    if (j == 0)              { w = fabsf(s_theta[1]);                      arg = vol; }
    else if (j == MESH - 1)  { w = fabsf(s_theta[tid]);                    arg = vol * vol; }
    else                     { w = fabsf(s_theta[tid] - s_theta[tid + 1]); arg = vol; }
    float contrib = w * __expf(-arg / avg_len);
    block_reduce2(contrib, 0.0f, s_w0, nullptr, &ws[2], nullptr);
}

__global__ void ssh_init_ws(double* __restrict__ ws) {
    if (threadIdx.x < 3) ws[threadIdx.x] = 0.0;
}

__global__ void ssh_finalize(const double* __restrict__ ws, float* __restrict__ out) {
    out[0] = (float)(ws[0] + ws[2] + ws[1]);   // loss_v + loss_so2 + loss_s
}

extern "C" void kernel_launch(void* const* d_in, const int* in_sizes, int n_in,
                              void* d_out, int out_size, void* d_ws, size_t ws_size,
                              hipStream_t stream) {
    const float*  theta = (const float*)d_in[0];   // MESH-1 floats
    const float2* state = (const float2*)d_in[1];  // MESH x 2 floats
    double* ws  = (double*)d_ws;
    float*  out = (float*)d_out;
    const int nblk = MESH / BLK;

    hipLaunchKernelGGL(ssh_init_ws,  dim3(1),    dim3(32),  0, stream, ws);
    hipLaunchKernelGGL(ssh_pass1,    dim3(nblk), dim3(BLK), 0, stream, theta, state, ws);
    hipLaunchKernelGGL(ssh_pass2,    dim3(nblk), dim3(BLK), 0, stream, theta, state, ws);
    hipLaunchKernelGGL(ssh_finalize, dim3(1),    dim3(1),   0, stream, ws, out);
}